// CauchyKernel_37374805409880
// MI455X (gfx1250) — compile-verified
//
#include <hip/hip_runtime.h>
#include <stdint.h>

// Problem constants (from reference)
#define SEQ_LEN   4096
#define D_MODEL   1024
#define NUM_POLES 64

// Tiling
#define D_TILE 128                 // d columns per block == blockDim.x
#define S_TILE 32                  // s rows per block (per-thread accumulators)
#define ROW_DW 66                  // 64 data dwords + 2 pad dwords (bank-conflict-free)
#define TILE_DW   (D_TILE * ROW_DW)   // 8448 dwords per array
#define TILE_BYTES (TILE_DW * 4)      // 33792 bytes per array

typedef unsigned int u32x4_t __attribute__((ext_vector_type(4)));
typedef int          i32x4_t __attribute__((ext_vector_type(4)));
typedef int          i32x8_t __attribute__((ext_vector_type(8)));

// Issue a TDM load of a (NUM_POLES x D_TILE) f32 tile (row-major, row stride
// NUM_POLES) from global memory into LDS at byte offset `lds_byte_off`,
// padding each 64-dword row by 2 dwords (LDS row stride = 66 dwords).
static __device__ __forceinline__ void tdm_load_tile(const float* gsrc,
                                                     uint32_t lds_byte_off) {
  uint64_t ga = (uint64_t)(uintptr_t)gsrc;

  // ---- D# group 0 (128b): count/lds_addr/global_addr/type ----
  u32x4_t g0;
  g0.x = 1u;                                   // count=1 (valid), user mode
  g0.y = lds_byte_off;                         // lds_addr (bytes)
  g0.z = (uint32_t)ga;                         // global_addr[31:0]
  g0.w = (uint32_t)((ga >> 32) & 0x1FFFFFFull) // global_addr[56:32]
       | (2u << 30);                           // type = 2 ("image")

  // ---- D# group 1 (256b): sizes/strides/padding ----
  i32x8_t g1;
  g1[0] = (int)((2u << 16)      // data_size = 4 bytes
              | (1u << 20)      // pad_enable
              | (5u << 22)      // pad_interval enc 5 -> every 64 dwords
              | (1u << 25));    // pad_amount enc 1 -> 2 dwords
  g1[1] = (int)((uint32_t)NUM_POLES << 16);    // tensor_dim0 = 64
  g1[2] = (int)((uint32_t)D_TILE    << 16);    // tensor_dim1 = 128
  g1[3] = (int)((uint32_t)NUM_POLES << 16);    // tile_dim0   = 64
  g1[4] = (int)((uint32_t)D_TILE);             // tile_dim1=128, tile_dim2=0
  g1[5] = NUM_POLES;                           // tensor_dim0_stride = 64
  g1[6] = 0;                                   // stride0 hi / stride1 lo
  g1[7] = 0;                                   // stride1 hi

  i32x4_t gz = {0, 0, 0, 0};                   // groups 2/3: unused (2D tile)

#if __has_include(<hip/amd_detail/amd_gfx1250_TDM.h>)
  // amdgpu-toolchain (clang-23): 6-arg form
  i32x8_t gz8 = {0, 0, 0, 0, 0, 0, 0, 0};
  __builtin_amdgcn_tensor_load_to_lds(g0, g1, gz, gz, gz8, 0);
#else
  // ROCm 7.2 (clang-22): 5-arg form
  __builtin_amdgcn_tensor_load_to_lds(g0, g1, gz, gz, 0);
#endif
}

extern "C" __global__ void __launch_bounds__(D_TILE)
cauchy_sum_kernel(const float* __restrict__ z,
                  const float* __restrict__ poles,
                  const float* __restrict__ residues,
                  float* __restrict__ out) {
  extern __shared__ float smem[];       // [0, TILE_DW) poles, [TILE_DW, 2*TILE_DW) residues

  const int t  = threadIdx.x;           // d-offset within tile: 0..127
  const int d0 = blockIdx.x * D_TILE;
  const int s0 = blockIdx.y * S_TILE;

  // Wave 0 issues both TDM tile loads and waits for completion; the
  // workgroup barrier then releases the remaining waves.
  if (t < 32) {
    tdm_load_tile(poles    + (size_t)d0 * NUM_POLES, 0u);
    tdm_load_tile(residues + (size_t)d0 * NUM_POLES, (uint32_t)TILE_BYTES);
    __builtin_amdgcn_s_wait_tensorcnt(0);
  }
  __syncthreads();

  // z values for this s-tile (uniform across the block -> scalar loads).
  float zv[S_TILE];
#pragma unroll
  for (int i = 0; i < S_TILE; ++i) zv[i] = z[s0 + i];

  float acc[S_TILE];
#pragma unroll
  for (int i = 0; i < S_TILE; ++i) acc[i] = 0.0f;

  // Per-thread row pointers; LDS bank for (t, p) is (2t + p) mod 64 -> no conflicts.
  const float* __restrict__ lp = smem + (size_t)t * ROW_DW;
  const float* __restrict__ lr = smem + TILE_DW + (size_t)t * ROW_DW;

#pragma unroll 4
  for (int p = 0; p < NUM_POLES; ++p) {
    const float pole = lp[p];
    const float res  = lr[p];
#pragma unroll
    for (int i = 0; i < S_TILE; ++i) {
      float x = zv[i] - pole;
      float r = __builtin_amdgcn_rcpf(x);               // v_rcp_f32 (TRANS pipe)
      float e = __builtin_fmaf(-x, r, 1.0f);            // Newton refinement
      r = __builtin_fmaf(r, e, r);                      // -> ~0.5 ulp division
      acc[i] = __builtin_fmaf(res, r, acc[i]);
    }
  }

  // Coalesced stores: lane-contiguous in d for each s row.
  float* o = out + (size_t)s0 * D_MODEL + d0 + t;
#pragma unroll
  for (int i = 0; i < S_TILE; ++i) o[(size_t)i * D_MODEL] = acc[i];
}

extern "C" void kernel_launch(void* const* d_in, const int* in_sizes, int n_in,
                              void* d_out, int out_size, void* d_ws, size_t ws_size,
                              hipStream_t stream) {
  (void)in_sizes; (void)n_in; (void)out_size; (void)d_ws; (void)ws_size;
  const float* z        = (const float*)d_in[0];
  const float* poles    = (const float*)d_in[1];
  const float* residues = (const float*)d_in[2];
  float*       out      = (float*)d_out;

  dim3 grid(D_MODEL / D_TILE, SEQ_LEN / S_TILE);   // (8, 128)
  dim3 block(D_TILE);                               // 128 threads = 4 waves
  size_t smem_bytes = 2 * (size_t)TILE_BYTES;       // 67,584 B of LDS
  hipLaunchKernelGGL(cauchy_sum_kernel, grid, block, smem_bytes, stream,
                     z, poles, residues, out);
}